// LuongAttention_50594714747318
// MI455X (gfx1250) — compile-verified
//
#include <hip/hip_runtime.h>
#include <math.h>

// ---------------------------------------------------------------------------
// Luong attention, MI455X (gfx1250, wave32).
// Roofline: h_j = 512MB dominates; kernel 1 reads it exactly once (online
// softmax + context accumulation, NT loads). Kernels 2/3 touch only ~4MB.
// Projection GEMM uses V_WMMA_F32_16X16X4_F32.
// ---------------------------------------------------------------------------

typedef float v2f __attribute__((ext_vector_type(2)));
typedef float v4f __attribute__((ext_vector_type(4)));
typedef float v8f __attribute__((ext_vector_type(8)));

constexpr int B  = 32;
constexpr int L  = 8192;
constexpr int D  = 512;
constexpr int K2 = 2 * D;          // 1024 (concat dim)
constexpr int NC = 32;             // chunks along L
constexpr int ROWS = L / NC;       // 256 rows per workgroup
constexpr int TPB  = 256;          // 8 waves (wave32)
constexpr int WAVES = TPB / 32;

// ---------------------------------------------------------------------------
// Kernel 1: fused scores + online softmax partials + partial context.
// One workgroup = (batch b, chunk c). Each wave owns interleaved rows.
// Per lane: 16 contiguous-after-interleave floats of the D=512 row, loaded as
// 4x float4 with layout d = j*128 + lane*4 (each b128 load is a contiguous
// 512B stream across the wave).
// ---------------------------------------------------------------------------
__global__ __launch_bounds__(TPB)
void k_scores_partials(const float* __restrict__ h,
                       const float* __restrict__ s,
                       float* __restrict__ e_buf,      // [B*L] raw scores
                       float* __restrict__ m_part,     // [B*NC]
                       float* __restrict__ s_part,     // [B*NC]
                       float* __restrict__ ctx_part) { // [B*NC*D]
  const int b     = blockIdx.x / NC;
  const int chunk = blockIdx.x % NC;
  const int lane  = threadIdx.x & 31;
  const int wave  = threadIdx.x >> 5;

  __shared__ float lm[WAVES];
  __shared__ float ls[WAVES];
  __shared__ float lctx[D];
  for (int i = threadIdx.x; i < D; i += TPB) lctx[i] = 0.0f;

  // preload s_i slice for this lane (same for every row of this batch)
  v4f sv[4];
#pragma unroll
  for (int j = 0; j < 4; ++j)
    sv[j] = *(const v4f*)(s + (size_t)b * D + j * 128 + lane * 4);

  float m  = -1e30f;
  float sa = 0.0f;
  v4f cx[4];
#pragma unroll
  for (int j = 0; j < 4; ++j) cx[j] = (v4f){0.f, 0.f, 0.f, 0.f};

  for (int it = 0; it < ROWS / WAVES; ++it) {
    const int l = chunk * ROWS + it * WAVES + wave;
    const float* hp = h + ((size_t)b * L + l) * D;
    v4f hv[4];
#pragma unroll
    for (int j = 0; j < 4; ++j)   // streaming, zero reuse -> non-temporal
      hv[j] = __builtin_nontemporal_load((const v4f*)(hp + j * 128 + lane * 4));

    v4f pv = hv[0] * sv[0] + hv[1] * sv[1] + hv[2] * sv[2] + hv[3] * sv[3];
    float p = pv.x + pv.y + pv.z + pv.w;
#pragma unroll
    for (int off = 16; off > 0; off >>= 1)  // wave32 butterfly reduce
      p += __shfl_xor(p, off, 32);
    // p == e_{b,l} on all lanes
    if (lane == 0) e_buf[(size_t)b * L + l] = p;

    const float mn   = fmaxf(m, p);
    const float corr = __expf(m - mn);
    const float w    = __expf(p - mn);
    sa = sa * corr + w;
#pragma unroll
    for (int j = 0; j < 4; ++j) cx[j] = cx[j] * corr + w * hv[j];
    m = mn;
  }

  __syncthreads();                       // lctx zeroed, all waves done
  if (lane == 0) { lm[wave] = m; ls[wave] = sa; }
  __syncthreads();

  float mg = lm[0];
#pragma unroll
  for (int wv = 1; wv < WAVES; ++wv) mg = fmaxf(mg, lm[wv]);
  const float scale = __expf(m - mg);    // this wave's rescale factor

#pragma unroll
  for (int j = 0; j < 4; ++j) {
    v4f t = cx[j] * scale;
#pragma unroll
    for (int c = 0; c < 4; ++c)
      atomicAdd(&lctx[j * 128 + lane * 4 + c], t[c]);  // ds_add_f32
  }
  __syncthreads();

  const int rec = b * NC + chunk;
  if (threadIdx.x == 0) {
    float sg = 0.0f;
#pragma unroll
    for (int wv = 0; wv < WAVES; ++wv) sg += ls[wv] * __expf(lm[wv] - mg);
    m_part[rec] = mg;
    s_part[rec] = sg;
  }
  for (int i = threadIdx.x; i < D; i += TPB)
    ctx_part[(size_t)rec * D + i] = lctx[i];
}

// ---------------------------------------------------------------------------
// Kernel 2: per-batch merge of chunk records; writes normalized attention
// weights `a` and the tanh'd concat vector u = tanh([ctx, s_i]).
// Only ~4MB of traffic total; 32 blocks.
// ---------------------------------------------------------------------------
__global__ __launch_bounds__(256)
void k_combine(const float* __restrict__ e_buf,
               const float* __restrict__ m_part,
               const float* __restrict__ s_part,
               const float* __restrict__ ctx_part,
               const float* __restrict__ s_i,
               float* __restrict__ a_out,   // [B*L]
               float* __restrict__ u) {     // [B*2D]
  const int b = blockIdx.x;

  float mg = -1e30f;
  for (int c = 0; c < NC; ++c) mg = fmaxf(mg, m_part[b * NC + c]);
  float sg = 0.0f;
  for (int c = 0; c < NC; ++c)
    sg += s_part[b * NC + c] * __expf(m_part[b * NC + c] - mg);
  const float inv = 1.0f / sg;

  for (int l = threadIdx.x; l < L; l += 256)
    a_out[(size_t)b * L + l] = __expf(e_buf[(size_t)b * L + l] - mg) * inv;

  for (int d = threadIdx.x; d < D; d += 256) {
    float acc = 0.0f;
    for (int c = 0; c < NC; ++c)
      acc += __expf(m_part[b * NC + c] - mg) *
             ctx_part[(size_t)(b * NC + c) * D + d];
    u[(size_t)b * K2 + d]     = tanhf(acc * inv);
    u[(size_t)b * K2 + D + d] = tanhf(s_i[(size_t)b * D + d]);
  }
}

// ---------------------------------------------------------------------------
// Kernel 3: att[32,512] = U[32,1024] x W^T + bias via V_WMMA_F32_16X16X4_F32.
// 2 M-tiles x 32 N-tiles = 64 waves, one 16x16 tile each; K-loop of 256 WMMAs.
// Fragment layouts per ISA 7.12.2:
//   A 16x4 : lane = M (0..15) in each half; K = 2*(lane>>4) + vgpr
//   B 4x16 : lane = N (0..15) in each half; K = 2*(lane>>4) + vgpr
//   C/D    : vgpr r -> M = r + 8*(lane>>4), N = lane&15
// ---------------------------------------------------------------------------
__global__ __launch_bounds__(128)
void k_proj_wmma(const float* __restrict__ u,
                 const float* __restrict__ Wp,     // [D, 2D] row-major (out,in)
                 const float* __restrict__ bias,   // [D]
                 float* __restrict__ att) {        // [B, D]
  const int wave = (blockIdx.x * blockDim.x + threadIdx.x) >> 5;  // 0..63
  const int lane = threadIdx.x & 31;
  const int mt = wave >> 5;        // 0..1
  const int nt = wave & 31;        // 0..31
  const int hi = lane >> 4;        // lane half
  const int lo = lane & 15;

  const float* ua = u  + (size_t)(mt * 16 + lo) * K2;  // A row (M = lo)
  const float* wb = Wp + (size_t)(nt * 16 + lo) * K2;  // B col n = W row n

  v8f acc = {0.f, 0.f, 0.f, 0.f, 0.f, 0.f, 0.f, 0.f};
  for (int k0 = 0; k0 < K2; k0 += 4) {
    const int kk = k0 + 2 * hi;
    v2f av = *(const v2f*)(ua + kk);
    v2f bv = *(const v2f*)(wb + kk);
    acc = __builtin_amdgcn_wmma_f32_16x16x4_f32(
        /*neg_a=*/false, av, /*neg_b=*/false, bv,
        /*c_mod=*/(short)0, acc, /*reuse_a=*/false, /*reuse_b=*/false);
  }

  const int rowb = mt * 16 + hi * 8;
  const int col  = nt * 16 + lo;
  const float bc = bias[col];
#pragma unroll
  for (int r = 0; r < 8; ++r)
    att[(size_t)(rowb + r) * D + col] = acc[r] + bc;
}

// ---------------------------------------------------------------------------
extern "C" void kernel_launch(void* const* d_in, const int* in_sizes, int n_in,
                              void* d_out, int out_size, void* d_ws,
                              size_t ws_size, hipStream_t stream) {
  (void)in_sizes; (void)n_in; (void)out_size; (void)ws_size;

  const float* h_j    = (const float*)d_in[0];  // [B, L, D]
  const float* s_i    = (const float*)d_in[1];  // [B, 1, D]
  const float* W_proj = (const float*)d_in[2];  // [D, 2D]
  const float* b_proj = (const float*)d_in[3];  // [D]

  float* out   = (float*)d_out;
  float* a_out = out;                       // [B, L]
  float* att   = out + (size_t)B * L;       // [B, D]

  // workspace layout (floats)
  float* ws       = (float*)d_ws;
  float* e_buf    = ws;                                   // B*L
  float* m_part   = e_buf + (size_t)B * L;                // B*NC
  float* s_part   = m_part + B * NC;                      // B*NC
  float* ctx_part = s_part + B * NC;                      // B*NC*D
  float* u        = ctx_part + (size_t)B * NC * D;        // B*2D

  k_scores_partials<<<dim3(B * NC), dim3(TPB), 0, stream>>>(
      h_j, s_i, e_buf, m_part, s_part, ctx_part);
  k_combine<<<dim3(B), dim3(256), 0, stream>>>(
      e_buf, m_part, s_part, ctx_part, s_i, a_out, u);
  k_proj_wmma<<<dim3(16), dim3(128), 0, stream>>>(u, W_proj, b_proj, att);
}